// TransformGtClassification2x1d_57303453663898
// MI455X (gfx1250) — compile-verified
//
#include <hip/hip_runtime.h>
#include <stdint.h>

// Problem constants (match reference)
#define WBINS 1024
#define KSIZE 31
#define KHALF 15

typedef float vf4 __attribute__((ext_vector_type(4)));
typedef int   v4i __attribute__((vector_size(16)));   // matches builtin param

#if defined(__has_builtin)
#if __has_builtin(__builtin_amdgcn_global_store_async_from_lds_b128)
#define USE_ASYNC_LDS_STORE 1
#endif
#endif

#if USE_ASYNC_LDS_STORE
typedef __attribute__((address_space(1))) v4i gvec_t;  // "__device__" pointer
typedef __attribute__((address_space(3))) v4i lvec_t;  // "__shared__" pointer
#endif

// ---------------------------------------------------------------------------
// out[row, j] = gauss[idx_row - j + 15] when 0 <= idx_row - j + 15 <= 30,
// else 0 (conv of one-hot with 31-tap kernel == stamp kernel at idx).
//
// One block (8 wave32) per 1024-float row; thread t owns columns 4t..4t+3 and
// emits exactly one 16-byte store. 256 MiB of output streams at the HBM store
// floor (~11.5 us @ 23.3 TB/s); compute is a handful of VALU ops per element.
//
// CDNA5 path: stage the 4 KiB row in LDS, then per-lane
// GLOBAL_STORE_ASYNC_FROM_LDS_B128 (ASYNCcnt) streams it to memory; drain
// with s_wait_asynccnt. Fallback: direct non-temporal b128 stores (output
// exceeds the 192 MB L2 and is never re-read).
// ---------------------------------------------------------------------------
__global__ __launch_bounds__(256) void row_scatter(
    const int* __restrict__ gt, const float* __restrict__ gauss,
    float* __restrict__ out) {
    const int t = threadIdx.x;
    const int row = blockIdx.x;          // uniform -> scalar load of gt[row]
    int idx = gt[row];
    idx = (idx > WBINS - 1) ? (WBINS - 1) : idx;

    const int j0 = t << 2;
    vf4 v = {0.0f, 0.0f, 0.0f, 0.0f};
    const int u0 = idx - j0 + KHALF;     // tap index for column j0
    if (u0 >= 0 && u0 <= KSIZE + 2) {    // any of u0..u0-3 lands in [0,30]
#pragma unroll
        for (int e = 0; e < 4; ++e) {
            const unsigned u = (unsigned)(u0 - e);  // wraps huge when negative
            if (u <= (unsigned)(KSIZE - 1)) v[e] = gauss[u];  // L2-resident taps
        }
    }

#if USE_ASYNC_LDS_STORE
    __shared__ vf4 stage[256];           // one full row: 4 KiB
    stage[t] = v;
    __syncthreads();                     // DS writes visible before async read

    gvec_t* gp = (gvec_t*)(uintptr_t)(out + (size_t)row * WBINS + j0);
    // Generic LDS pointer: low 32 bits are the LDS byte offset (aperture rule)
    lvec_t* lp = (lvec_t*)(uint32_t)(uintptr_t)(&stage[t]);

    __builtin_amdgcn_global_store_async_from_lds_b128(gp, lp, 0, 0);

#if __has_builtin(__builtin_amdgcn_s_wait_asynccnt)
    __builtin_amdgcn_s_wait_asynccnt(0);
#else
    asm volatile("s_wait_asynccnt 0x0" ::: "memory");
#endif
#else
    __builtin_nontemporal_store(v, (vf4*)(out + (size_t)row * WBINS + j0));
#endif
}

// ---------------------------------------------------------------------------
// Launch: inputs are gt (int32 [B,2] flat = 65536) and gauss1d (float [31]).
// Output: float32 [B,2,W] flat. One block per output row.
// ---------------------------------------------------------------------------
extern "C" void kernel_launch(void* const* d_in, const int* in_sizes, int n_in,
                              void* d_out, int out_size, void* d_ws, size_t ws_size,
                              hipStream_t stream) {
    const int*   gt    = (const int*)d_in[0];
    const float* gauss = (const float*)d_in[1];
    float*       out   = (float*)d_out;

    const int rows = in_sizes[0];        // B*2 = 65536
    dim3 grid((unsigned)rows), block(256);
    hipLaunchKernelGGL(row_scatter, grid, block, 0, stream, gt, gauss, out);
}